// GCN_11845519802407
// MI455X (gfx1250) — compile-verified
//
#include <hip/hip_runtime.h>
#include <hip/hip_bf16.h>

typedef float v2f __attribute__((ext_vector_type(2)));
typedef float v8f __attribute__((ext_vector_type(8)));

// ---------------- degree / symmetric norm ----------------
__global__ void k_deg_init(float* deg, int n) {
  int i = blockIdx.x * blockDim.x + threadIdx.x;
  if (i < n) deg[i] = 1.0f;  // self-loop contributes 1 to every node's degree
}

__global__ void k_deg_accum(const int* __restrict__ dst, float* deg, int e) {
  int i = blockIdx.x * blockDim.x + threadIdx.x;
  if (i < e) atomicAdd(&deg[dst[i]], 1.0f);
}

__global__ void k_deg_rsqrt(float* deg, int n) {
  int i = blockIdx.x * blockDim.x + threadIdx.x;
  if (i < n) deg[i] = rsqrtf(deg[i]);  // deg >= 1 always
}

// ---------------- fp32 GEMM via V_WMMA_F32_16X16X4_F32 ----------------
// H[N,Kout] = X[N,Kin] @ W[Kin,Kout].  One wave per 16x16 output tile.
// W is staged in LDS once per block; K-loop fully unrolled (compile-time shapes).
template <int Kin, int Kout>
__global__ void k_gemm_wmma(const float* __restrict__ X, const float* __restrict__ W,
                            float* __restrict__ H, int N, int tiles) {
  __shared__ float ldsW[Kin * Kout];  // <= 32 KB, LDS/WGP is 320 KB
  for (int i = threadIdx.x; i < Kin * Kout; i += blockDim.x) ldsW[i] = W[i];
  __syncthreads();

  const int lane = threadIdx.x & 31;
  const int wave = threadIdx.x >> 5;
  const int tileId = blockIdx.x * (blockDim.x >> 5) + wave;
  if (tileId >= tiles) return;  // wave-uniform: EXEC stays all-ones for WMMA

  constexpr int colTiles = Kout >> 4;
  const int rowTile = tileId / colTiles;
  const int colTile = tileId - rowTile * colTiles;

  const int half = lane >> 4;  // 0: lanes 0-15, 1: lanes 16-31
  const int l16  = lane & 15;

  // A fragment: row M = l16, K = k + 2*half + {0,1}  (two adjacent floats -> b64 load)
  const float* xrow = X + (size_t)(rowTile * 16 + l16) * Kin + half * 2;
  // B fragment from LDS: VGPR0 -> K = k+2*half, VGPR1 -> K+1, N-col = colTile*16 + l16
  const float* wl = ldsW + (size_t)half * 2 * Kout + colTile * 16 + l16;

  v8f c = {};
#pragma unroll
  for (int k = 0; k < Kin; k += 4) {
    if ((k & 15) == 0 && k + 16 < Kin)
      __builtin_prefetch(xrow + k + 16, 0, 0);  // global_prefetch_b8
    v2f a = {xrow[k], xrow[k + 1]};
    v2f b = {wl[(size_t)k * Kout], wl[(size_t)(k + 1) * Kout]};
    // 8 args: (neg_a, A, neg_b, B, c_mod, C, reuse_a, reuse_b)
    c = __builtin_amdgcn_wmma_f32_16x16x4_f32(false, a, false, b, (short)0, c,
                                              false, false);
  }

  // C/D layout: VGPR r -> M = r + 8*half, N = l16
  float* hbase = H + (size_t)(rowTile * 16) * Kout + colTile * 16 + l16;
#pragma unroll
  for (int r = 0; r < 8; ++r) {
    const int m = r + half * 8;
    hbase[(size_t)m * Kout] = c[r];
  }
}

// ---------------- aggregation (float4-vectorized) ----------------
// agg = h * dinv^2   (self-loop message)
template <int F>
__global__ void k_self_init(const float* __restrict__ h, const float* __restrict__ dinv,
                            float* __restrict__ agg, int n) {
  constexpr int C = F / 4;
  int idx = blockIdx.x * blockDim.x + threadIdx.x;
  if (idx >= n * C) return;
  int node = idx / C;  // C is a power of two -> shift
  float di = dinv[node];
  float s = di * di;
  float4 v = ((const float4*)h)[idx];
  float4 o = {v.x * s, v.y * s, v.z * s, v.w * s};
  ((float4*)agg)[idx] = o;
}

// agg[dst] += h[src] * dinv[src]*dinv[dst]   per edge, per 4-feature chunk
template <int F>
__global__ void k_edge_scatter(const int* __restrict__ src, const int* __restrict__ dst,
                               const float* __restrict__ h, const float* __restrict__ dinv,
                               float* agg, int e) {
  constexpr int C = F / 4;
  int idx = blockIdx.x * blockDim.x + threadIdx.x;
  if (idx >= e * C) return;
  int ed = idx / C;
  int ch = idx & (C - 1);
  int s = src[ed], d = dst[ed];
  float norm = dinv[s] * dinv[d];
  float4 hv = *(const float4*)(h + (size_t)s * F + ch * 4);  // coalesced b128 gather
  float* base = agg + (size_t)d * F + ch * 4;
  atomicAdd(base + 0, hv.x * norm);
  atomicAdd(base + 1, hv.y * norm);
  atomicAdd(base + 2, hv.z * norm);
  atomicAdd(base + 3, hv.w * norm);
}

// x = relu(agg + b), in place
template <int F>
__global__ void k_bias_relu(float* agg, const float* __restrict__ b, int n) {
  constexpr int C = F / 4;
  int idx = blockIdx.x * blockDim.x + threadIdx.x;
  if (idx >= n * C) return;
  int ch = idx & (C - 1);
  float4 bv = *(const float4*)(b + ch * 4);
  float4 v = ((const float4*)agg)[idx];
  float4 o = {fmaxf(v.x + bv.x, 0.0f), fmaxf(v.y + bv.y, 0.0f),
              fmaxf(v.z + bv.z, 0.0f), fmaxf(v.w + bv.w, 0.0f)};
  ((float4*)agg)[idx] = o;
}

// ---------------- pooling + FC + log_softmax ----------------
__global__ void k_zero(float* p, int n) {
  int i = blockIdx.x * blockDim.x + threadIdx.x;
  if (i < n) p[i] = 0.0f;
}

__global__ void k_pool_accum(const float* __restrict__ x, const int* __restrict__ batch,
                             float* sums, float* cnts, int n) {
  // one thread per (node, 4-feature chunk), F = 64 -> 16 chunks
  int idx = blockIdx.x * blockDim.x + threadIdx.x;
  if (idx >= n * 16) return;
  int node = idx >> 4;
  int ch = idx & 15;
  int g = batch[node];
  float4 v = ((const float4*)x)[idx];
  float* base = sums + (size_t)g * 64 + ch * 4;
  atomicAdd(base + 0, v.x);
  atomicAdd(base + 1, v.y);
  atomicAdd(base + 2, v.z);
  atomicAdd(base + 3, v.w);
  if (ch == 0) atomicAdd(&cnts[g], 1.0f);
}

__global__ void k_pool_fc_lsm(const float* __restrict__ sums, const float* __restrict__ cnts,
                              const float* __restrict__ fcw, const float* __restrict__ fcb,
                              float* __restrict__ out, int G) {
  int g = blockIdx.x * blockDim.x + threadIdx.x;
  if (g >= G) return;
  float inv = 1.0f / fmaxf(cnts[g], 1.0f);
  float logits[6];
#pragma unroll
  for (int j = 0; j < 6; ++j) logits[j] = fcb[j];
  for (int f = 0; f < 64; ++f) {
    float p = sums[(size_t)g * 64 + f] * inv;
#pragma unroll
    for (int j = 0; j < 6; ++j) logits[j] += p * fcw[f * 6 + j];
  }
  float m = logits[0];
#pragma unroll
  for (int j = 1; j < 6; ++j) m = fmaxf(m, logits[j]);
  float s = 0.0f;
#pragma unroll
  for (int j = 0; j < 6; ++j) s += expf(logits[j] - m);
  float lse = m + logf(s);
#pragma unroll
  for (int j = 0; j < 6; ++j) out[(size_t)g * 6 + j] = logits[j] - lse;
}

// ---------------- host launch ----------------
extern "C" void kernel_launch(void* const* d_in, const int* in_sizes, int n_in,
                              void* d_out, int out_size, void* d_ws, size_t ws_size,
                              hipStream_t stream) {
  const float* x     = (const float*)d_in[0];
  const int*   ei    = (const int*)d_in[1];
  const int*   batch = (const int*)d_in[2];
  const float* W1 = (const float*)d_in[3];
  const float* b1 = (const float*)d_in[4];
  const float* W2 = (const float*)d_in[5];
  const float* b2 = (const float*)d_in[6];
  const float* W3 = (const float*)d_in[7];
  const float* b3 = (const float*)d_in[8];
  const float* fcw = (const float*)d_in[9];
  const float* fcb = (const float*)d_in[10];
  float* out = (float*)d_out;

  const int N = in_sizes[0] / 128;
  const int E = in_sizes[1] / 2;
  const int G = out_size / 6;
  const int* src = ei;
  const int* dst = ei + E;

  float* ws   = (float*)d_ws;
  float* dinv = ws;                      // N
  float* H    = dinv + N;                // N*128 (GEMM output)
  float* AGG  = H + (size_t)N * 128;     // N*128 (aggregation / layer output)
  float* sums = AGG + (size_t)N * 128;   // G*64
  float* cnts = sums + (size_t)G * 64;   // G (contiguous with sums)

  const int BT = 256;  // 8 wave32s per block
  auto cdiv = [](long long a, long long b) { return (int)((a + b - 1) / b); };

  // symmetric normalization coefficients (same for all 3 layers)
  k_deg_init <<<cdiv(N, BT), BT, 0, stream>>>(dinv, N);
  k_deg_accum<<<cdiv(E, BT), BT, 0, stream>>>(dst, dinv, E);
  k_deg_rsqrt<<<cdiv(N, BT), BT, 0, stream>>>(dinv, N);

#define RUN_LAYER(XIN, KIN, KOUT, W, B)                                                  \
  {                                                                                      \
    const int tiles = ((N + 15) >> 4) * (KOUT >> 4);                                     \
    k_gemm_wmma<KIN, KOUT><<<cdiv(tiles, BT >> 5), BT, 0, stream>>>(XIN, W, H, N, tiles);\
    const int nc = N * (KOUT / 4);                                                       \
    const int ec = E * (KOUT / 4);                                                       \
    k_self_init<KOUT>   <<<cdiv(nc, BT), BT, 0, stream>>>(H, dinv, AGG, N);              \
    k_edge_scatter<KOUT><<<cdiv(ec, BT), BT, 0, stream>>>(src, dst, H, dinv, AGG, E);    \
    k_bias_relu<KOUT>   <<<cdiv(nc, BT), BT, 0, stream>>>(AGG, B, N);                    \
  }

  RUN_LAYER(x,   128,  64, W1, b1);   // -> AGG (N x 64)
  RUN_LAYER(AGG,  64, 128, W2, b2);   // -> AGG (N x 128)
  RUN_LAYER(AGG, 128,  64, W3, b3);   // -> AGG (N x 64)
#undef RUN_LAYER

  // global mean pool + FC + log_softmax
  k_zero      <<<cdiv(G * 65, BT), BT, 0, stream>>>(sums, G * 65);  // sums + cnts
  k_pool_accum<<<cdiv((long long)N * 16, BT), BT, 0, stream>>>(AGG, batch, sums, cnts, N);
  k_pool_fc_lsm<<<cdiv(G, 64), 64, 0, stream>>>(sums, cnts, fcw, fcb, out, G);
}